// QAEPlusAncCRY_40458591928897
// MI455X (gfx1250) — compile-verified
//
#include <hip/hip_runtime.h>

// QAE circuit: exploit linearity. Phase 1 builds the fixed 512x256 complex
// transfer matrix M = U*S by simulating the circuit on S's 256 columns
// (one wave32 per column, state in VGPRs, cross-lane gates via shuffles).
// Phase 2 computes Y = X * M^T (re & im) with V_WMMA_F32_16X16X4_F32 and
// fuses the |y|^2 group-reduction + affine output.

#define INVSQRT2 0.70710678118654752f

typedef __attribute__((ext_vector_type(2))) float v2f;
typedef __attribute__((ext_vector_type(8))) float v8f;

// ---------------- Phase 1: gate helpers ----------------
// Flat index i (9 bits): bits 8..4 = lane (5 bits), bits 3..0 = k (reg array).
// Qubit q acts on bit T = 8 - q. Reference applies G^T (tensordot convention):
//   RY:  a0' = c*a0 + s*a1 ; a1' = -s*a0 + c*a1
//   RZ:  diag(e^{-ia/2}, e^{+ia/2})  (diagonal -> always lane-local)
//   CNOT: symmetric swap.

template<int T>
__device__ __forceinline__ void ry_bit(float (&re)[16], float (&im)[16],
                                       int lane, float c, float s) {
  if constexpr (T < 4) {
    #pragma unroll
    for (int k = 0; k < 16; ++k) {
      if ((k & (1 << T)) == 0) {
        const int k1 = k | (1 << T);
        float r0 = re[k], r1 = re[k1];
        re[k]  = c * r0 + s * r1;
        re[k1] = c * r1 - s * r0;
        float i0 = im[k], i1 = im[k1];
        im[k]  = c * i0 + s * i1;
        im[k1] = c * i1 - s * i0;
      }
    }
  } else {
    const int mask = 1 << (T - 4);
    const float se = (lane & mask) ? -s : s;   // bit-clear side gets +s
    #pragma unroll
    for (int k = 0; k < 16; ++k) {
      float orr = __shfl_xor(re[k], mask, 32);
      float ori = __shfl_xor(im[k], mask, 32);
      re[k] = c * re[k] + se * orr;
      im[k] = c * im[k] + se * ori;
    }
  }
}

template<int T>
__device__ __forceinline__ void rz_bit(float (&re)[16], float (&im)[16],
                                       int lane, float c, float s) {
  if constexpr (T < 4) {
    #pragma unroll
    for (int k = 0; k < 16; ++k) {
      const float sp = ((k >> T) & 1) ? s : -s;
      float r = re[k], i = im[k];
      re[k] = c * r - sp * i;
      im[k] = c * i + sp * r;
    }
  } else {
    const float sp = ((lane >> (T - 4)) & 1) ? s : -s;
    #pragma unroll
    for (int k = 0; k < 16; ++k) {
      float r = re[k], i = im[k];
      re[k] = c * r - sp * i;
      im[k] = c * i + sp * r;
    }
  }
}

template<int TC, int TT>
__device__ __forceinline__ void cnot_bits(float (&re)[16], float (&im)[16],
                                          int lane) {
  if constexpr (TC < 4 && TT < 4) {
    #pragma unroll
    for (int k = 0; k < 16; ++k) {
      if ((k & (1 << TC)) && !(k & (1 << TT))) {
        const int k1 = k | (1 << TT);
        float t = re[k]; re[k] = re[k1]; re[k1] = t;
        t = im[k]; im[k] = im[k1]; im[k1] = t;
      }
    }
  } else if constexpr (TC < 4) {  // control in k, target in lane bits
    const int mask = 1 << (TT - 4);
    #pragma unroll
    for (int k = 0; k < 16; ++k) {
      float orr = __shfl_xor(re[k], mask, 32);
      float ori = __shfl_xor(im[k], mask, 32);
      if (k & (1 << TC)) { re[k] = orr; im[k] = ori; }
    }
  } else if constexpr (TT < 4) {  // control in lane bits, target in k
    const bool ctrl = (lane >> (TC - 4)) & 1;
    #pragma unroll
    for (int k = 0; k < 16; ++k) {
      if (!(k & (1 << TT))) {
        const int k1 = k | (1 << TT);
        float a = re[k], b = re[k1];
        re[k] = ctrl ? b : a; re[k1] = ctrl ? a : b;
        float ai = im[k], bi = im[k1];
        im[k] = ctrl ? bi : ai; im[k1] = ctrl ? ai : bi;
      }
    }
  } else {                        // both in lane bits
    const int mask = 1 << (TT - 4);
    const bool ctrl = (lane >> (TC - 4)) & 1;
    #pragma unroll
    for (int k = 0; k < 16; ++k) {
      float orr = __shfl_xor(re[k], mask, 32);
      float ori = __shfl_xor(im[k], mask, 32);
      if (ctrl) { re[k] = orr; im[k] = ori; }
    }
  }
}

template<int TT>  // CRY: control = bit 0 (ancilla), target bit TT in {1,2,3}
__device__ __forceinline__ void cry_bit(float (&re)[16], float (&im)[16],
                                        float c, float s) {
  #pragma unroll
  for (int k = 0; k < 16; ++k) {
    if ((k & 1) && !(k & (1 << TT))) {
      const int k1 = k | (1 << TT);
      float r0 = re[k], r1 = re[k1];
      re[k]  = c * r0 + s * r1;
      re[k1] = c * r1 - s * r0;
      float i0 = im[k], i1 = im[k1];
      im[k]  = c * i0 + s * i1;
      im[k1] = c * i1 - s * i0;
    }
  }
}

__device__ __forceinline__ void ry_q(int q, float (&re)[16], float (&im)[16],
                                     int lane, float c, float s) {
  switch (q) {
    case 0: ry_bit<8>(re, im, lane, c, s); break;
    case 1: ry_bit<7>(re, im, lane, c, s); break;
    case 2: ry_bit<6>(re, im, lane, c, s); break;
    case 3: ry_bit<5>(re, im, lane, c, s); break;
    case 4: ry_bit<4>(re, im, lane, c, s); break;
    case 5: ry_bit<3>(re, im, lane, c, s); break;
    case 6: ry_bit<2>(re, im, lane, c, s); break;
    default: ry_bit<1>(re, im, lane, c, s); break;
  }
}

__device__ __forceinline__ void rz_q(int q, float (&re)[16], float (&im)[16],
                                     int lane, float c, float s) {
  switch (q) {
    case 0: rz_bit<8>(re, im, lane, c, s); break;
    case 1: rz_bit<7>(re, im, lane, c, s); break;
    case 2: rz_bit<6>(re, im, lane, c, s); break;
    case 3: rz_bit<5>(re, im, lane, c, s); break;
    case 4: rz_bit<4>(re, im, lane, c, s); break;
    case 5: rz_bit<3>(re, im, lane, c, s); break;
    case 6: rz_bit<2>(re, im, lane, c, s); break;
    default: rz_bit<1>(re, im, lane, c, s); break;
  }
}

__device__ __forceinline__ void cnot_q(int tc, int tt, float (&re)[16],
                                       float (&im)[16], int lane) {
  switch ((tc << 4) | tt) {
    case 0x87: cnot_bits<8,7>(re, im, lane); break;
    case 0x76: cnot_bits<7,6>(re, im, lane); break;
    case 0x65: cnot_bits<6,5>(re, im, lane); break;
    case 0x54: cnot_bits<5,4>(re, im, lane); break;
    case 0x43: cnot_bits<4,3>(re, im, lane); break;
    case 0x32: cnot_bits<3,2>(re, im, lane); break;
    case 0x21: cnot_bits<2,1>(re, im, lane); break;
    case 0x18: cnot_bits<1,8>(re, im, lane); break;
    case 0x81: cnot_bits<8,1>(re, im, lane); break;
    case 0x78: cnot_bits<7,8>(re, im, lane); break;
    case 0x67: cnot_bits<6,7>(re, im, lane); break;
    case 0x56: cnot_bits<5,6>(re, im, lane); break;
    case 0x45: cnot_bits<4,5>(re, im, lane); break;
    case 0x34: cnot_bits<3,4>(re, im, lane); break;
    case 0x23: cnot_bits<2,3>(re, im, lane); break;
    case 0x12: cnot_bits<1,2>(re, im, lane); break;
    default: break;
  }
}

// ---------------- Phase 1 kernel: build M (layout [part][j][i]) ----------------
__global__ __launch_bounds__(256) void qae_build_M(
    const float* __restrict__ th_ry, const float* __restrict__ th_rz,
    const float* __restrict__ th_cry, float* __restrict__ Mg) {
  const int lane = threadIdx.x & 31;
  const int j = blockIdx.x * 8 + (threadIdx.x >> 5);  // column 0..255

  float re[16], im[16];
  const int jh = j >> 4, jl = j & 15;
  #pragma unroll
  for (int k = 0; k < 16; ++k) {
    re[k] = (((lane & 15) == jh) && (k == jl)) ? INVSQRT2 : 0.0f;
    im[k] = 0.0f;
  }

  int i_rz = 0;
  for (int m = 0; m < 50; ++m) {
    const int base = m * 8;
    #pragma unroll
    for (int q = 0; q < 8; ++q) {
      float a = 0.5f * th_ry[base + q];
      ry_q(q, re, im, lane, __cosf(a), __sinf(a));
    }
    if (((m + 1) % 5) == 0) {
      #pragma unroll
      for (int q = 0; q < 8; ++q) {
        float a = 0.5f * th_rz[i_rz + q];
        rz_q(q, re, im, lane, __cosf(a), __sinf(a));
      }
      i_rz += 8;
    }
    const bool fwd = ((m & 1) == 0);
    #pragma unroll
    for (int q = 0; q < 8; ++q) {
      const int tq = fwd ? ((q + 1) & 7) : ((q + 7) & 7);
      cnot_q(8 - q, 8 - tq, re, im, lane);
    }
  }
  #pragma unroll
  for (int q = 0; q < 8; ++q) {
    float a = 0.5f * th_ry[400 + q];
    ry_q(q, re, im, lane, __cosf(a), __sinf(a));
  }
  {
    float a = 0.5f * th_cry[0]; cry_bit<3>(re, im, __cosf(a), __sinf(a));
    a = 0.5f * th_cry[1];       cry_bit<2>(re, im, __cosf(a), __sinf(a));
    a = 0.5f * th_cry[2];       cry_bit<1>(re, im, __cosf(a), __sinf(a));
  }

  // Store M[part][j][i], i = lane*16 + k  (coalesced: 16 consecutive floats/lane)
  float* r0 = Mg + (size_t)j * 512 + lane * 16;
  float* r1 = Mg + 131072 + (size_t)j * 512 + lane * 16;
  #pragma unroll
  for (int k = 0; k < 16; ++k) { r0[k] = re[k]; r1[k] = im[k]; }
}

// ---------------- Phase 2: f32 WMMA GEMM + fused probability ----------------
#define XSTR 258  // padded LDS row stride (floats): conflict-free ds_load_b64

__global__ __launch_bounds__(256) void qae_gemm_prob(
    const float* __restrict__ X, const float* __restrict__ Mg,
    const float* __restrict__ beta_p, const float* __restrict__ bias,
    float* __restrict__ out) {
  __shared__ float Xs[16 * XSTR];
  __shared__ float Ps[16][8];
  const int tid = threadIdx.x;
  const int bt  = blockIdx.x;  // batch tile (16 rows)

  // Stage X tile (16 x 256) into LDS
  #pragma unroll
  for (int i = 0; i < 16; ++i) {
    const int e = tid + i * 256;
    const int r = e >> 8, c = e & 255;
    Xs[r * XSTR + c] = X[(bt * 16 + r) * 256 + c];
  }
  if (tid < 128) Ps[tid >> 3][tid & 7] = 0.0f;

  // Warm L2/WGP$ with M (global_prefetch_b8)
  for (int off = tid; off < 2048; off += 256)
    __builtin_prefetch(Mg + (size_t)off * 128, 0, 1);

  __syncthreads();

  const int lane = tid & 31;
  const int wave = tid >> 5;
  const int row  = lane & 15;          // A-row m / B-col n for this lane
  const int koff = (lane >> 4) * 2;    // K sub-offset within each K=4 step

  float pacc[8];
  #pragma unroll
  for (int v = 0; v < 8; ++v) pacc[v] = 0.0f;

  // 64 tasks per block: part(re/im) x 32 A-slices; 8 per wave
  for (int task = wave; task < 64; task += 8) {
    const int part  = task >> 5;
    const int slice = task & 31;
    const float* Mp = Mg + part * 131072 + slice * 16 + row;  // + kk*512
    v8f acc = {0.f, 0.f, 0.f, 0.f, 0.f, 0.f, 0.f, 0.f};
    for (int kb = 0; kb < 256; kb += 4) {
      const int kk = kb + koff;
      v2f a, b;
      const float* xs = &Xs[row * XSTR + kk];
      a.x = xs[0];
      a.y = xs[1];
      b.x = Mp[(size_t)kk * 512];
      b.y = Mp[(size_t)(kk + 1) * 512];
      acc = __builtin_amdgcn_wmma_f32_16x16x4_f32(
          false, a, false, b, (short)0, acc, false, false);
    }
    #pragma unroll
    for (int v = 0; v < 8; ++v) pacc[v] += acc[v] * acc[v];
  }

  // Lane holds column n=row of i=slice*16+n: class = bits 3..1 of n,
  // ancilla = bit 0 (summed implicitly). m = v + 8*(lane>>4).
  const int ng = row >> 1;
  const int mh = (lane >> 4) * 8;
  #pragma unroll
  for (int v = 0; v < 8; ++v)
    atomicAdd(&Ps[mh + v][ng], pacc[v]);

  __syncthreads();

  if (tid < 128) {
    const int m = tid >> 3, n = tid & 7;
    const float beta = beta_p[0];
    const float p = Ps[m][n];
    out[(bt * 16 + m) * 8 + n] = beta * (2.0f * p - 1.0f) + bias[n];
  }
}

// ---------------- Host entry ----------------
extern "C" void kernel_launch(void* const* d_in, const int* in_sizes, int n_in,
                              void* d_out, int out_size, void* d_ws, size_t ws_size,
                              hipStream_t stream) {
  (void)in_sizes; (void)n_in; (void)out_size; (void)ws_size;
  const float* x      = (const float*)d_in[0];
  const float* th_ry  = (const float*)d_in[1];
  const float* th_rz  = (const float*)d_in[2];
  const float* th_cry = (const float*)d_in[3];
  const float* beta   = (const float*)d_in[4];
  const float* bias   = (const float*)d_in[5];
  float* Mg  = (float*)d_ws;   // 2 * 512 * 256 floats = 1 MB scratch
  float* out = (float*)d_out;

  qae_build_M<<<32, 256, 0, stream>>>(th_ry, th_rz, th_cry, Mg);
  qae_gemm_prob<<<64, 256, 0, stream>>>(x, Mg, beta, bias, out);
}